// Paillier_31585189495142
// MI455X (gfx1250) — compile-verified
//
#include <hip/hip_runtime.h>

// The Python harness owns main(), I/O, device malloc/memcpy, and timing.

typedef float v4f __attribute__((ext_vector_type(4)));

// Exact fmod(a, 225) for a in [0, 3600) — pure f32 path, branchless.
// q = trunc(a/225) estimated via reciprocal is off by at most +-1
// (abs err <= 16*2^-23 << 1). The intermediate fma may round when q is
// wrong, but rounding is monotone around the grid points 0 and 225, so the
// select decisions are still correct; the final fma with corrected q rounds
// the exact remainder, which needs <= 24 significant bits -> bit-exact fmod.
__device__ __forceinline__ float fmod225_small(float a) {
    float q = __builtin_truncf(__fmul_rn(a, (1.0f / 225.0f)));
    float r = __builtin_fmaf(q, -225.0f, a);
    float adj = (r < 0.0f) ? -1.0f : ((r >= 225.0f) ? 1.0f : 0.0f);
    return __builtin_fmaf(q + adj, -225.0f, a);   // exact
}

// Exact fmod(a, 225) for any f32 a in [0, 2^59) — f64 path, branchless.
// a converts exactly to double; q <= 1.95e15 < 2^53; q estimate via
// reciprocal has abs err ~0.43 < 1 -> off by at most +-1. The intermediate
// fma is *exact* in f64 even with the off-by-one q (for a >= 2^24, a is an
// integer so r is an integer in (-225,450); for a < 2^24 everything fits),
// so we can correct r directly with +-225 (also exact) — no second fma.
// The true remainder (< 225, a multiple of ulp(a_f32)) is exactly
// representable in f32. Bit-exact with C fmodf.
__device__ __forceinline__ float fmod225_big(float af) {
    double a = (double)af;
    double q = __builtin_trunc(a * (1.0 / 225.0));
    double r = __builtin_fma(q, -225.0, a);       // exact
    r = (r < 0.0)    ? (r + 225.0) : r;           // exact
    r = (r >= 225.0) ? (r - 225.0) : r;           // exact
    return (float)r;                               // exact
}

__device__ __forceinline__ float paillier_elem(float x, float r) {
    // nude = fmod(15*x + 1, 225); 15x+1 in [1,16) so the fmod is identity.
    // __fmul_rn/__fadd_rn block fma-contraction (reference rounds mul and
    // add separately).
    float nude = __fadd_rn(__fmul_rn(15.0f, x), 1.0f);

    // r^15 with lax.integer_pow's square-and-multiply rounding order:
    // acc = ((r * r^2) * r^4) * r^8
    float s1 = __fmul_rn(r, r);    // r^2
    float a1 = __fmul_rn(r, s1);   // r^3
    float s2 = __fmul_rn(s1, s1);  // r^4
    float a2 = __fmul_rn(a1, s2);  // r^7
    float s3 = __fmul_rn(s2, s2);  // r^8
    float p15 = __fmul_rn(a2, s3); // r^15  (< 4.4e17, finite)

    float obf = fmod225_big(p15);                 // [0, 225)
    float prod = __fmul_rn(nude, obf);            // < 16*225 = 3600
    return fmod225_small(prod);
}

__global__ __launch_bounds__(256) void Paillier_31585189495142_kernel(
    const v4f* __restrict__ xv, const v4f* __restrict__ rv,
    v4f* __restrict__ ov, unsigned nvec, unsigned ntail) {
    const unsigned tid    = blockIdx.x * blockDim.x + threadIdx.x;
    const unsigned stride = gridDim.x * blockDim.x;

    // Streaming main loop: 128-bit non-temporal loads/stores (805 MB >> 192 MB
    // L2, each byte touched once) + gfx1250 global_prefetch_b8 one grid-stride
    // ahead (speculative; OOB prefetches are silently dropped). 32-bit element
    // indexing (buffer is 256 MB) keeps addressing in SADDR+voffset form.
    for (unsigned i = tid; i < nvec; i += stride) {
        __builtin_prefetch(&xv[i + stride], 0, 0);
        __builtin_prefetch(&rv[i + stride], 0, 0);

        v4f xq = __builtin_nontemporal_load(&xv[i]);
        v4f rq = __builtin_nontemporal_load(&rv[i]);

        v4f o;
        o.x = paillier_elem(xq.x, rq.x);
        o.y = paillier_elem(xq.y, rq.y);
        o.z = paillier_elem(xq.z, rq.z);
        o.w = paillier_elem(xq.w, rq.w);

        __builtin_nontemporal_store(o, &ov[i]);
    }

    // Scalar tail (n % 4 == 0 for this problem; kept for generality).
    if (ntail) {
        const float* xp = (const float*)xv;
        const float* rp = (const float*)rv;
        float* op       = (float*)ov;
        for (unsigned j = nvec * 4u + tid; j < nvec * 4u + ntail; j += stride) {
            op[j] = paillier_elem(xp[j], rp[j]);
        }
    }
}

extern "C" void kernel_launch(void* const* d_in, const int* in_sizes, int n_in,
                              void* d_out, int out_size, void* d_ws, size_t ws_size,
                              hipStream_t stream) {
    (void)in_sizes; (void)n_in; (void)d_ws; (void)ws_size;

    const v4f* x = (const v4f*)d_in[0];  // plaintext, f32
    const v4f* r = (const v4f*)d_in[1];  // r_value,  f32
    v4f* out     = (v4f*)d_out;

    const unsigned n     = (unsigned)out_size;   // 64*1024*1024
    const unsigned nvec  = n >> 2;
    const unsigned ntail = n & 3u;

    const int block = 256;                       // 8 wave32 waves per block
    unsigned blocksNeeded = (nvec + block - 1) / block;
    if (blocksNeeded < 1) blocksNeeded = 1;
    // Cap at 16384 blocks -> ~4 float4 iterations/thread: enough resident
    // waves to saturate HBM, and the +stride prefetch is useful for 3/4 of
    // the iterations.
    int grid = blocksNeeded > 16384u ? 16384 : (int)blocksNeeded;

    Paillier_31585189495142_kernel<<<grid, block, 0, stream>>>(x, r, out, nvec, ntail);
}